// linear_H_48069273977167
// MI455X (gfx1250) — compile-verified
//
#include <hip/hip_runtime.h>
#include <math.h>

#define D   128
#define D2  256

typedef float v2f __attribute__((ext_vector_type(2)));
typedef float v8f __attribute__((ext_vector_type(8)));

// ---------------------------------------------------------------------------
// Kernel 0: zero agg + scalar accumulator (contiguous in workspace)
// ---------------------------------------------------------------------------
__global__ __launch_bounds__(256) void zero_kernel(float* __restrict__ p, long long n) {
    long long i = (long long)blockIdx.x * blockDim.x + threadIdx.x;
    long long stride = (long long)gridDim.x * blockDim.x;
    for (; i < n; i += stride) p[i] = 0.0f;
}

// ---------------------------------------------------------------------------
// Kernel 1: h = sin(x @ W1^T + b1) via V_WMMA_F32_16X16X4_F32
//   block = 256 threads = 8 waves; block covers a 32-row strip of x (LDS, 32KB),
//   wave w computes the 32x16 output tile at columns [16w, 16w+16):
//   two 16x16 C tiles (rows lo/hi) sharing each B fragment -> 2 WMMA per W1 load.
// ---------------------------------------------------------------------------
__global__ __launch_bounds__(256) void gemm1_sin_kernel(
    const float* __restrict__ x,    // [N, 256] row-major
    const float* __restrict__ W1,   // [128, 256] row-major (out_feat, in_feat)
    const float* __restrict__ bias1,// [128]
    float* __restrict__ h,          // [N, 128]
    int n_rows)
{
    __shared__ float xs[32 * D2];   // 32 KB strip of x

    const int tid  = threadIdx.x;
    const int row0 = blockIdx.x * 32;

    // Cooperative LDS stage: 32 rows x 256 floats = 2048 float4, 256 threads x 8.
    {
        const float4* xg = (const float4*)x;
        float4*       xl = (float4*)xs;
#pragma unroll
        for (int i = 0; i < 8; ++i) {
            int idx = tid + i * 256;          // float4 index within strip
            int r   = row0 + (idx >> 6);      // 64 float4 per row
            float4 v = {0.f, 0.f, 0.f, 0.f};
            if (r < n_rows) v = xg[(long long)r * (D2 / 4) + (idx & 63)];
            xl[idx] = v;
        }
    }
    __syncthreads();

    const int wave = tid >> 5;        // 0..7  -> column tile
    const int lane = tid & 31;
    const int half = lane >> 4;       // 0: K pair (k,k+1), 1: K pair (k+2,k+3)
    const int l15  = lane & 15;
    const int col0 = wave * 16;

    // A 16x4 f32 layout: lanes 0-15 hold (M=l15, K=kk..kk+1), lanes 16-31 (M=l15, K=kk+2..kk+3)
    // B 4x16 f32 layout (mirrored): lanes 0-15 hold (N=l15, K=kk..kk+1), lanes 16-31 K=kk+2..kk+3
    const float* alo  = xs + l15 * D2;                        // rows row0 + 0..15
    const float* ahi  = xs + (16 + l15) * D2;                 // rows row0 + 16..31
    const float* brow = W1 + (long long)(col0 + l15) * D2;    // W1[col][k] = B[k][col]

    v8f clo0 = {}; v8f clo1 = {}; v8f chi0 = {}; v8f chi1 = {};

#pragma unroll 4
    for (int kk = 0; kk < D2; kk += 8) {
        const int kb = kk + 2 * half;
        v2f a, b0, b1;

        b0.x = brow[kb + 0]; b0.y = brow[kb + 1];
        b1.x = brow[kb + 4]; b1.y = brow[kb + 5];

        a.x = alo[kb + 0]; a.y = alo[kb + 1];
        clo0 = __builtin_amdgcn_wmma_f32_16x16x4_f32(false, a, false, b0, (short)0, clo0, false, false);
        a.x = ahi[kb + 0]; a.y = ahi[kb + 1];
        chi0 = __builtin_amdgcn_wmma_f32_16x16x4_f32(false, a, false, b0, (short)0, chi0, false, false);
        a.x = alo[kb + 4]; a.y = alo[kb + 5];
        clo1 = __builtin_amdgcn_wmma_f32_16x16x4_f32(false, a, false, b1, (short)0, clo1, false, false);
        a.x = ahi[kb + 4]; a.y = ahi[kb + 5];
        chi1 = __builtin_amdgcn_wmma_f32_16x16x4_f32(false, a, false, b1, (short)0, chi1, false, false);
    }

    v8f clo = clo0 + clo1;
    v8f chi = chi0 + chi1;

    // Epilogue: C/D layout — VGPR v holds row M = v + 8*half, col = col0 + l15.
    const float bias = bias1[col0 + l15];
#pragma unroll
    for (int v = 0; v < 8; ++v) {
        const int m   = v + 8 * half;
        const int rlo = row0 + m;
        const int rhi = row0 + 16 + m;
        if (rlo < n_rows) h[(long long)rlo * D + col0 + l15] = sinf(clo[v] + bias);
        if (rhi < n_rows) h[(long long)rhi * D + col0 + l15] = sinf(chi[v] + bias);
    }
}

// ---------------------------------------------------------------------------
// Kernel 2: SpMM scatter-add.  32 lanes per edge, float4 per lane.
//   agg[row[e]] += w[e] * h[col[e]]   (L2-resident gather + native fp32 atomics)
// ---------------------------------------------------------------------------
__global__ __launch_bounds__(256) void spmm_kernel(
    const long long* __restrict__ row_idx,
    const long long* __restrict__ col_idx,
    const float* __restrict__ ew,
    const float* __restrict__ h,
    float* __restrict__ agg,
    int nedges)
{
    const int e    = blockIdx.x * 8 + (threadIdx.x >> 5);
    const int lane = threadIdx.x & 31;
    if (e >= nedges) return;

    const long long r = row_idx[e];
    const long long c = col_idx[e];
    const float     w = ew[e];

    const float4 v = ((const float4*)(h + (long long)c * D))[lane];
    float* ag = agg + (long long)r * D + lane * 4;

    unsafeAtomicAdd(ag + 0, w * v.x);
    unsafeAtomicAdd(ag + 1, w * v.y);
    unsafeAtomicAdd(ag + 2, w * v.z);
    unsafeAtomicAdd(ag + 3, w * v.w);
}

// ---------------------------------------------------------------------------
// Kernel 3: out[r] = sin(dot(agg[r], W2) + b2); accumulate sum(out^2).
//   One wave per node (128 floats = 32 lanes x float4), shuffle reduce.
// ---------------------------------------------------------------------------
__global__ __launch_bounds__(256) void layer2_norm_kernel(
    const float* __restrict__ agg,
    const float* __restrict__ W2,   // [128]
    const float* __restrict__ b2,   // [1]
    float* __restrict__ acc,        // scalar accumulator
    int n)
{
    const int wave = threadIdx.x >> 5;
    const int lane = threadIdx.x & 31;
    const int r    = blockIdx.x * 8 + wave;

    float ssum = 0.0f;
    if (r < n) {
        const float4 a = ((const float4*)(agg + (long long)r * D))[lane];
        const float4 w = ((const float4*)W2)[lane];
        float d = a.x * w.x + a.y * w.y + a.z * w.z + a.w * w.w;
#pragma unroll
        for (int off = 16; off > 0; off >>= 1)
            d += __shfl_down(d, off, 32);
        if (lane == 0) {
            const float o = sinf(d + b2[0]);
            ssum = o * o;
        }
    }

    __shared__ float sacc[8];
    if (lane == 0) sacc[wave] = ssum;
    __syncthreads();
    if (threadIdx.x == 0) {
        float s = 0.0f;
#pragma unroll
        for (int i = 0; i < 8; ++i) s += sacc[i];
        unsafeAtomicAdd(acc, s);
    }
}

__global__ void finalize_kernel(const float* __restrict__ acc, float* __restrict__ out) {
    out[0] = sqrtf(acc[0]);
}

// ---------------------------------------------------------------------------
extern "C" void kernel_launch(void* const* d_in, const int* in_sizes, int n_in,
                              void* d_out, int out_size, void* d_ws, size_t ws_size,
                              hipStream_t stream)
{
    const float*     x    = (const float*)d_in[0];     // [N, 256]
    const float*     W1   = (const float*)d_in[1];     // [128, 256]
    const float*     b1   = (const float*)d_in[2];     // [128]
    const float*     W2   = (const float*)d_in[3];     // [1, 128]
    const float*     b2   = (const float*)d_in[4];     // [1]
    const long long* eidx = (const long long*)d_in[5]; // [2, E] int64
    const float*     ew   = (const float*)d_in[6];     // [E]

    const int N = in_sizes[0] / D2;   // 50000
    const int E = in_sizes[6];        // 800000

    // Workspace layout: h [N*D] | agg [N*D] | acc [1]
    float* h   = (float*)d_ws;
    float* agg = h + (long long)N * D;
    float* acc = agg + (long long)N * D;

    // zero agg + acc (contiguous) every call -> deterministic under graph replay
    const long long nz = (long long)N * D + 1;
    zero_kernel<<<1024, 256, 0, stream>>>(agg, nz);

    gemm1_sin_kernel<<<(N + 31) / 32, 256, 0, stream>>>(x, W1, b1, h, N);

    spmm_kernel<<<(E + 7) / 8, 256, 0, stream>>>(eidx, eidx + E, ew, h, agg, E);

    layer2_norm_kernel<<<(N + 7) / 8, 256, 0, stream>>>(agg, W2, b2, acc, N);

    finalize_kernel<<<1, 1, 0, stream>>>(acc, (float*)d_out);
}